// MedianConvolution_5377299054734
// MI455X (gfx1250) — compile-verified
//
#include <hip/hip_runtime.h>
#include <hip/hip_bf16.h>

// Problem constants (from reference)
#define NN     50000
#define KK     32
#define IN_C   256
#define OUT_C  128

typedef __attribute__((ext_vector_type(2))) float v2f;
typedef __attribute__((ext_vector_type(8))) float v8f;

// -----------------------------------------------------------------------------
// Kernel 1: h = x @ W  via V_WMMA_F32_16X16X4_F32 (full fp32 matrix path).
// One wave (32 lanes) computes one 16x16 tile of h. Block = 8 waves = one
// 16-row strip of x covering all 8 column tiles of W (A rows reused in cache).
//
// ISA VGPR layouts (CDNA5 ISA §7.12.2):
//   A 16x4 f32 : lanes 0-15 and 16-31 both hold M=0..15;
//                v0 = K0 (lanes 0-15) / K2 (lanes 16-31); v1 = K1 / K3.
//   B 4x16 f32 : rows striped across lanes; v0 = K0 row / K2 row; v1 = K1/K3.
//   C/D 16x16  : VGPR v holds row M=v (lanes 0-15) and M=v+8 (lanes 16-31).
// -----------------------------------------------------------------------------
__global__ void __launch_bounds__(256)
median_conv_gemm_f32(const float* __restrict__ x,
                     const float* __restrict__ w,
                     float* __restrict__ h) {
    const int wave = threadIdx.x >> 5;          // 0..7
    const int lane = threadIdx.x & 31;
    const int lid  = lane & 15;                 // matrix row / col within tile
    const int half = lane >> 4;                 // 0: K{0,1}, 1: K{2,3}

    const int gw = blockIdx.x * 8 + wave;       // global wave-tile id, 0..24999
    const int mt = gw >> 3;                     // 0..3124  (50000/16 = 3125)
    const int nt = gw & 7;                      // 0..7     (128/16 = 8)
    const int m0 = mt * 16;
    const int n0 = nt * 16;

    const float* __restrict__ xrow = x + (size_t)(m0 + lid) * IN_C;

    v8f c = {};
    #pragma unroll 4
    for (int k0 = 0; k0 < IN_C; k0 += 4) {
        v2f a, b;
        const int ka = k0 + 2 * half;
        a.x = xrow[ka];
        a.y = xrow[ka + 1];
        b.x = w[(size_t)ka       * OUT_C + n0 + lid];
        b.y = w[(size_t)(ka + 1) * OUT_C + n0 + lid];
        // 8 args: (neg_a, A, neg_b, B, c_mod, C, reuse_a, reuse_b)
        c = __builtin_amdgcn_wmma_f32_16x16x4_f32(
                false, a, false, b, (short)0, c, false, false);
    }

    #pragma unroll
    for (int v = 0; v < 8; v++) {
        h[(size_t)(m0 + v + 8 * half) * OUT_C + n0 + lid] = c[v];
    }
}

// -----------------------------------------------------------------------------
// Kernel 2: per-node, per-channel lower median of up to 32 neighbor values.
// Block = 128 threads = one node; thread = channel c, so each neighbor gather
// h[nb*128 + c] is a coalesced 512B row read (h is L2-resident: 25.6MB << 192MB).
// Padding slots (j >= deg) become +inf, matching the reference sort semantics.
// Fully-unrolled 32-element bitonic sort keeps everything in VGPRs; the
// median index is block-uniform, extracted by a cndmask select tree (no
// runtime register-array indexing -> no scratch spill).
// -----------------------------------------------------------------------------
__global__ void __launch_bounds__(128)
median_conv_select(const float* __restrict__ h,
                   const int*   __restrict__ nbr,
                   const int*   __restrict__ deg,
                   float*       __restrict__ out) {
    const int n = blockIdx.x;                   // node
    const int c = threadIdx.x;                  // channel 0..127

    const int d   = deg[n];                     // 1..32, block-uniform
    const int med = (d - 1) >> 1;               // lower-median index, uniform

    const int* __restrict__ nb = nbr + (size_t)n * KK;

    float v[KK];
    #pragma unroll
    for (int j = 0; j < KK; j++) {
        const int idx = nb[j];                  // uniform -> scalar load
        const float val = h[(size_t)idx * OUT_C + c];
        v[j] = (j < d) ? val : __builtin_inff();
    }

    // Bitonic sort, ascending; all control folds at compile time.
    #pragma unroll
    for (int k = 2; k <= KK; k <<= 1) {
        #pragma unroll
        for (int j = k >> 1; j > 0; j >>= 1) {
            #pragma unroll
            for (int i = 0; i < KK; i++) {
                const int l = i ^ j;
                if (l > i) {
                    const bool up = ((i & k) == 0);
                    const float a = v[i], b = v[l];
                    const float lo = fminf(a, b);
                    const float hi = fmaxf(a, b);
                    v[i] = up ? lo : hi;
                    v[l] = up ? hi : lo;
                }
            }
        }
    }

    // Uniform-index extraction: med'th smallest.
    float r = v[0];
    #pragma unroll
    for (int i = 1; i < KK; i++) {
        r = (med == i) ? v[i] : r;
    }

    out[(size_t)n * OUT_C + c] = r;
}

// -----------------------------------------------------------------------------
// Launch: inputs in setup_inputs() order: x[f32], kernel[f32], neighbors[i32],
// degrees[i32]. Output: [50000,128] f32. Workspace holds h (25.6 MB).
// -----------------------------------------------------------------------------
extern "C" void kernel_launch(void* const* d_in, const int* in_sizes, int n_in,
                              void* d_out, int out_size, void* d_ws, size_t ws_size,
                              hipStream_t stream) {
    const float* x   = (const float*)d_in[0];
    const float* w   = (const float*)d_in[1];
    const int*   nbr = (const int*)d_in[2];
    const int*   deg = (const int*)d_in[3];
    float*       out = (float*)d_out;
    float*       h   = (float*)d_ws;            // NN * OUT_C * 4 = 25.6 MB

    // GEMM: 25000 wave-tiles, 8 waves/block -> 3125 blocks of 256 threads.
    median_conv_gemm_f32<<<3125, 256, 0, stream>>>(x, w, h);

    // Median: one 128-thread block per node.
    median_conv_select<<<NN, 128, 0, stream>>>(h, nbr, deg, out);
}